// Block_76673756168658
// MI455X (gfx1250) — compile-verified
//
#include <hip/hip_runtime.h>
#include <hip/hip_bf16.h>
#include <math.h>

// ---------------------------------------------------------------------------
// Transformer block: pre-LN attention (flash-style, fp32 WMMA) + pre-LN FFN.
// B=32, T=1024, C=32, H=4, d=8.
// ---------------------------------------------------------------------------

#define B_    32
#define T_    1024
#define C_    32
#define H_    4
#define D_    8
#define EPS_  1e-5f
#define SCALE_ 0.17677669529663687f   // 32^-0.5

typedef float v2f __attribute__((ext_vector_type(2)));
typedef float v8f __attribute__((ext_vector_type(8)));

// ===========================================================================
// Kernel 1: LN1 + QKV projection.  One thread per token.
// q/k/v written as [B,H,T,8] fp32.
// ===========================================================================
__global__ __launch_bounds__(256) void tb_qkv_kernel(
    const float* __restrict__ x,
    const float* __restrict__ Wq, const float* __restrict__ Wk,
    const float* __restrict__ Wv,
    const float* __restrict__ g1, const float* __restrict__ b1,
    float* __restrict__ qb, float* __restrict__ kb, float* __restrict__ vb)
{
    __shared__ float sWq[H_ * C_ * D_];   // 1024
    __shared__ float sWk[H_ * C_ * D_];
    __shared__ float sWv[H_ * C_ * D_];
    __shared__ float sg[C_], sb[C_];

    const int tid = threadIdx.x;
    for (int i = tid; i < H_ * C_ * D_; i += 256) {
        sWq[i] = Wq[i]; sWk[i] = Wk[i]; sWv[i] = Wv[i];
    }
    if (tid < C_) { sg[tid] = g1[tid]; sb[tid] = b1[tid]; }
    __syncthreads();

    const int t  = blockIdx.x * 256 + tid;        // global token, < B*T
    const int b  = t >> 10;
    const int tt = t & (T_ - 1);

    const float* xp = x + (size_t)t * C_;
    float h[C_];
    float mu = 0.f;
    #pragma unroll
    for (int c = 0; c < C_; c++) { h[c] = xp[c]; mu += h[c]; }
    mu *= (1.0f / C_);
    float var = 0.f;
    #pragma unroll
    for (int c = 0; c < C_; c++) { float d = h[c] - mu; var += d * d; }
    const float rs = rsqrtf(var * (1.0f / C_) + EPS_);
    #pragma unroll
    for (int c = 0; c < C_; c++) h[c] = (h[c] - mu) * rs * sg[c] + sb[c];

    for (int hh = 0; hh < H_; hh++) {
        const size_t base = ((size_t)(b * H_ + hh) * T_ + tt) * D_;
        const float* wq = sWq + hh * C_ * D_;
        const float* wk = sWk + hh * C_ * D_;
        const float* wv = sWv + hh * C_ * D_;
        float aq[D_] = {0,0,0,0,0,0,0,0};
        float ak[D_] = {0,0,0,0,0,0,0,0};
        float av[D_] = {0,0,0,0,0,0,0,0};
        for (int c = 0; c < C_; c++) {
            const float hv = h[c];
            #pragma unroll
            for (int d = 0; d < D_; d++) {
                aq[d] += hv * wq[c * D_ + d];
                ak[d] += hv * wk[c * D_ + d];
                av[d] += hv * wv[c * D_ + d];
            }
        }
        #pragma unroll
        for (int d = 0; d < D_; d++) {
            qb[base + d] = aq[d];
            kb[base + d] = ak[d];
            vb[base + d] = av[d];
        }
    }
}

// ===========================================================================
// Kernel 2: fused flash attention with V_WMMA_F32_16X16X4_F32.
// Grid: B*H*(T/64) blocks, 128 threads (4 waves).  Wave w owns 16 output rows
// t0 = rb*64 + w*16.  wei[t,s] = (k[t] . q[s]) * C^-0.5  (K rows = WMMA A,
// Q cols = WMMA B).  Online softmax over s, all reductions via the per-wave
// LDS score-tile bounce; l tracked as a "ones" column (n==8) of the P*V
// accumulator (V staged pre-padded to 16 columns).  Q/V chunks staged with
// GLOBAL_LOAD_ASYNC_TO_LDS_B128 (ASYNCcnt path, no VGPR round trip).
// attn written concatenated-heads: [B,T,C].
// ===========================================================================
#define SCHUNK 256

__global__ __launch_bounds__(128) void tb_attn_kernel(
    const float* __restrict__ qbuf, const float* __restrict__ kbuf,
    const float* __restrict__ vbuf, float* __restrict__ attnbuf)
{
    __shared__ float sQ[SCHUNK * D_];     // 8 KB
    __shared__ float sV[SCHUNK * 16];     // 16 KB: cols 0-7 = V, 8 = 1.0, 9-15 = 0
    __shared__ float sP[4][16 * 16];      // per-wave raw score tile
    __shared__ float sM[4][16];           // per-wave per-row softmax correction

    const int tid  = threadIdx.x;
    const int wave = tid >> 5;
    const int lane = tid & 31;
    const int ln   = lane & 15;        // N index / row index within strip
    const int hi   = lane >> 4;        // lane-half select

    const int idx = blockIdx.x;
    const int rb  = idx & 15;          // row block (T/64 = 16)
    const int bh  = idx >> 4;          // b*H + h
    const int hh  = bh & (H_ - 1);
    const int bb  = bh >> 2;

    const float* qh = qbuf + (size_t)bh * T_ * D_;
    const float* kh = kbuf + (size_t)bh * T_ * D_;
    const float* vh = vbuf + (size_t)bh * T_ * D_;

    const int t0 = rb * 64 + wave * 16;

    // --- A operand: K rows for this wave's 16-row strip (ISA A layout:
    //     lanes 0-15 rows M, VGPR pair = K 2*hi,2*hi+1).  Pre-scaled.
    v2f a0, a1;
    {
        const float* kp = kh + (size_t)(t0 + ln) * D_ + hi * 2;
        a0.x = kp[0] * SCALE_; a0.y = kp[1] * SCALE_;
        a1.x = kp[4] * SCALE_; a1.y = kp[5] * SCALE_;
    }

    // --- one-time init of the constant pad columns of sV: [.. | 1 | 0*7] ---
    for (int i = tid; i < SCHUNK * 2; i += 128) {
        const int row = i >> 1, g = i & 1;
        ((float4*)sV)[row * 4 + 2 + g] =
            (g == 0) ? make_float4(1.f, 0.f, 0.f, 0.f)
                     : make_float4(0.f, 0.f, 0.f, 0.f);
    }

    // LDS byte offsets (low 32 bits of the shared-aperture flat address)
    const uint32_t sQoff = (uint32_t)(uintptr_t)(&sQ[0]);
    const uint32_t sVoff = (uint32_t)(uintptr_t)(&sV[0]);

    v8f acc = {0.f,0.f,0.f,0.f,0.f,0.f,0.f,0.f};   // C layout; col 8 == l
    float mrun = -1e30f;                           // running max of row `ln`

    float* pp = sP[wave];
    float* pm = sM[wave];

    for (int c0 = 0; c0 < T_; c0 += SCHUNK) {
        __syncthreads();   // previous chunk fully consumed
        {
            const float* qsrc = qh + (size_t)c0 * D_;
            const float* vsrc = vh + (size_t)c0 * D_;
            #pragma unroll
            for (int k = 0; k < 4; k++) {
                const int i = tid + k * 128;              // 0..511
                // Q: contiguous 16B blocks
                const float*   qg   = qsrc + i * 4;
                const uint32_t qdst = sQoff + (uint32_t)i * 16u;
                asm volatile("global_load_async_to_lds_b128 %0, %1, off"
                             :: "v"(qdst), "v"(qg) : "memory");
                // V: 8-float rows scattered into 16-col padded layout
                const int row = i >> 1, g = i & 1;
                const float*   vg   = vsrc + row * 8 + g * 4;
                const uint32_t vdst = sVoff + (uint32_t)(row * 64 + g * 16);
                asm volatile("global_load_async_to_lds_b128 %0, %1, off"
                             :: "v"(vdst), "v"(vg) : "memory");
            }
            asm volatile("s_wait_asynccnt 0x0" ::: "memory");
        }
        __syncthreads();   // chunk visible to all waves

        for (int st = 0; st < SCHUNK; st += 16) {
            // ---- scores: S = (K * scale) @ Q^T  (2 WMMAs over d) ----
            v2f b0 = *(const v2f*)(sQ + (st + ln) * D_ + hi * 2);
            v2f b1 = *(const v2f*)(sQ + (st + ln) * D_ + 4 + hi * 2);
            v8f s = {0.f,0.f,0.f,0.f,0.f,0.f,0.f,0.f};
            s = __builtin_amdgcn_wmma_f32_16x16x4_f32(false, a0, false, b0,
                                                      (short)0, s, false, false);
            s = __builtin_amdgcn_wmma_f32_16x16x4_f32(false, a1, false, b1,
                                                      (short)0, s, false, false);

            // ---- raw score tile to per-wave LDS (C layout -> row major) ----
            #pragma unroll
            for (int r = 0; r < 8; r++)
                pp[(hi * 8 + r) * 16 + ln] = s[r];

            // ---- row max of row `ln`: contiguous b128 loads, VALU tree ----
            const float4* prow = (const float4*)(pp + ln * 16);
            const float4 q0 = prow[0], q1 = prow[1], q2 = prow[2], q3 = prow[3];
            float mt = fmaxf(fmaxf(fmaxf(q0.x, q0.y), fmaxf(q0.z, q0.w)),
                             fmaxf(fmaxf(q1.x, q1.y), fmaxf(q1.z, q1.w)));
            mt = fmaxf(mt, fmaxf(fmaxf(fmaxf(q2.x, q2.y), fmaxf(q2.z, q2.w)),
                                 fmaxf(fmaxf(q3.x, q3.y), fmaxf(q3.z, q3.w))));
            const float mn   = fmaxf(mrun, mt);
            const float corr = __expf(mrun - mn);
            mrun = mn;

            // ---- broadcast per-row correction, rescale accumulator ----
            if (lane < 16) pm[ln] = corr;
            #pragma unroll
            for (int r = 0; r < 8; r++)
                acc[r] *= pm[hi * 8 + r];

            // ---- P @ [V | 1 | 0...]: exp applied in A layout ----
            #pragma unroll
            for (int k = 0; k < 4; k++) {
                v2f pa = *(const v2f*)(pp + ln * 16 + k * 4 + hi * 2);
                pa.x = __expf(pa.x - mn);
                pa.y = __expf(pa.y - mn);
                const int srow = st + k * 4 + hi * 2;
                v2f vv;
                vv.x = sV[srow * 16 + ln];
                vv.y = sV[(srow + 1) * 16 + ln];
                acc = __builtin_amdgcn_wmma_f32_16x16x4_f32(false, pa, false, vv,
                                                            (short)0, acc, false, false);
            }
        }
    }

    // ---- finalize: divide by l (column 8 of acc), write [B,T,H*8] ----
    const int orow = t0 + hi * 8;
    #pragma unroll
    for (int r = 0; r < 8; r++) {
        const float lr = __shfl(acc[r], 8 + (lane & 16), 32);
        if (ln < 8) {
            attnbuf[((size_t)bb * T_ + orow + r) * C_ + hh * D_ + ln] = acc[r] / lr;
        }
    }
}

// ===========================================================================
// Kernel 3: out-proj + residual + LN2 + FFN(relu) + residual.
// One wave per token; 8 tokens per 256-thread block.  Weights in LDS,
// activations exchanged with lane shuffles.
// ===========================================================================
__global__ __launch_bounds__(256) void tb_ffn_kernel(
    const float* __restrict__ x,   const float* __restrict__ attn,
    const float* __restrict__ Wp,  const float* __restrict__ bp,
    const float* __restrict__ W1,  const float* __restrict__ b1,
    const float* __restrict__ W2,  const float* __restrict__ b2,
    const float* __restrict__ g2,  const float* __restrict__ be2,
    float* __restrict__ out)
{
    __shared__ float sWp[C_ * C_];        // 4 KB
    __shared__ float sW1[C_ * 4 * C_];    // 16 KB
    __shared__ float sW2[4 * C_ * C_];    // 16 KB
    __shared__ float sbp[C_], sb1[4 * C_], sb2[C_], sg[C_], sbe[C_];

    const int tid = threadIdx.x;
    for (int i = tid; i < C_ * C_; i += 256) sWp[i] = Wp[i];
    for (int i = tid; i < C_ * 4 * C_; i += 256) { sW1[i] = W1[i]; sW2[i] = W2[i]; }
    if (tid < C_) { sbp[tid] = bp[tid]; sb2[tid] = b2[tid]; sg[tid] = g2[tid]; sbe[tid] = be2[tid]; }
    if (tid < 4 * C_) sb1[tid] = b1[tid];
    __syncthreads();

    const int lane = tid & 31;
    const int t    = blockIdx.x * 8 + (tid >> 5);

    const float a  = attn[(size_t)t * C_ + lane];
    const float xv = x[(size_t)t * C_ + lane];

    // out-proj + bias + residual
    float x1 = sbp[lane] + xv;
    for (int c = 0; c < C_; c++)
        x1 += __shfl(a, c, 32) * sWp[c * C_ + lane];

    // LN2 (C_ == warp width == 32)
    float sm = x1;
    #pragma unroll
    for (int off = 16; off; off >>= 1) sm += __shfl_xor(sm, off, 32);
    const float mu = sm * (1.0f / C_);
    const float dv = x1 - mu;
    float s2 = dv * dv;
    #pragma unroll
    for (int off = 16; off; off >>= 1) s2 += __shfl_xor(s2, off, 32);
    const float h2 = dv * rsqrtf(s2 * (1.0f / C_) + EPS_) * sg[lane] + sbe[lane];

    // FFN up (128 hidden: lane j owns j, j+32, j+64, j+96) + relu
    float hid[4] = { sb1[lane], sb1[lane + 32], sb1[lane + 64], sb1[lane + 96] };
    for (int c = 0; c < C_; c++) {
        const float hc = __shfl(h2, c, 32);
        #pragma unroll
        for (int mq = 0; mq < 4; mq++)
            hid[mq] += hc * sW1[c * (4 * C_) + mq * 32 + lane];
    }
    #pragma unroll
    for (int mq = 0; mq < 4; mq++) hid[mq] = fmaxf(hid[mq], 0.f);

    // FFN down + bias + residual
    float o = sb2[lane] + x1;
    for (int mq = 0; mq < 4; mq++) {
        for (int c = 0; c < C_; c++) {
            const float hk = __shfl(hid[mq], c, 32);
            o += hk * sW2[(mq * 32 + c) * C_ + lane];
        }
    }
    out[(size_t)t * C_ + lane] = o;
}

// ===========================================================================
// Launch
// ===========================================================================
extern "C" void kernel_launch(void* const* d_in, const int* in_sizes, int n_in,
                              void* d_out, int out_size, void* d_ws, size_t ws_size,
                              hipStream_t stream)
{
    (void)in_sizes; (void)n_in; (void)out_size; (void)ws_size;

    const float* x     = (const float*)d_in[0];
    const float* Wq    = (const float*)d_in[1];
    const float* Wk    = (const float*)d_in[2];
    const float* Wv    = (const float*)d_in[3];
    const float* Wproj = (const float*)d_in[4];
    const float* bproj = (const float*)d_in[5];
    const float* W1    = (const float*)d_in[6];
    const float* b1    = (const float*)d_in[7];
    const float* W2    = (const float*)d_in[8];
    const float* b2    = (const float*)d_in[9];
    const float* ln1g  = (const float*)d_in[10];
    const float* ln1b  = (const float*)d_in[11];
    const float* ln2g  = (const float*)d_in[12];
    const float* ln2b  = (const float*)d_in[13];
    float* out = (float*)d_out;

    // workspace: q | k | v | attn, each B*H*T*8 == B*T*C == 1,048,576 floats
    const size_t headElems = (size_t)B_ * H_ * T_ * D_;
    float* qb = (float*)d_ws;
    float* kb = qb + headElems;
    float* vb = kb + headElems;
    float* ab = vb + headElems;

    // 1) LN1 + QKV
    tb_qkv_kernel<<<(B_ * T_) / 256, 256, 0, stream>>>(
        x, Wq, Wk, Wv, ln1g, ln1b, qb, kb, vb);

    // 2) fused WMMA flash attention (async-LDS staging)
    tb_attn_kernel<<<B_ * H_ * (T_ / 64), 128, 0, stream>>>(qb, kb, vb, ab);

    // 3) out-proj + LN2 + FFN
    tb_ffn_kernel<<<(B_ * T_) / 8, 256, 0, stream>>>(
        x, ab, Wproj, bproj, W1, b1, W2, b2, ln2g, ln2b, out);
}